// SpectralFeatureGraph_74955769249855
// MI455X (gfx1250) — compile-verified
//
#include <hip/hip_runtime.h>
#include <hip/hip_bf16.h>

// B=8, N=512, E=128, H=64
// K1: P[:,0:64]=h@W1 ; P[:,64:128]=h@W2+b   (fp32 WMMA 16x16x4)
// K2: e[b,j,k]=sum_h att[h]*LR(pj[j,h]+pk'[k,h]) ; alpha=softmax_k(e)
//     16 j-rows per WG, pk' staged via async global->LDS in h-chunks of 16.
// K3: edge_index(int64) + edge_weights = 0.5*(alpha+alpha^T) off-diagonal.

typedef float v2f __attribute__((ext_vector_type(2)));
typedef float v4f __attribute__((ext_vector_type(4)));
typedef float v8f __attribute__((ext_vector_type(8)));
typedef int   v4i __attribute__((vector_size(4 * sizeof(int))));

#define BB   8
#define NN   512
#define EE   128
#define HH   64
#define MROWS (BB * NN)          // 4096
#define BE1  (NN * (NN - 1))     // 261632 edges per batch
#define BE   (BB * BE1)          // 2093056 edges total

// ---- CDNA5 async global->LDS copy (ASYNCcnt-tracked), with asm fallback ----
typedef __attribute__((address_space(1))) v4i* gptr_b128;
typedef __attribute__((address_space(3))) v4i* lptr_b128;

__device__ __forceinline__ void async_copy_b128(const float* g, float* l) {
#if __has_builtin(__builtin_amdgcn_global_load_async_to_lds_b128)
  __builtin_amdgcn_global_load_async_to_lds_b128(
      (gptr_b128)(void*)g, (lptr_b128)(void*)l, 0, 0);
#else
  unsigned           llo = (unsigned)(unsigned long long)(void*)l; // LDS = low 32 bits
  unsigned long long ga  = (unsigned long long)(const void*)g;
  asm volatile("global_load_async_to_lds_b128 %0, %1, off"
               :: "v"(llo), "v"(ga) : "memory");
#endif
}
__device__ __forceinline__ void wait_async0() {
#if __has_builtin(__builtin_amdgcn_s_wait_asynccnt)
  __builtin_amdgcn_s_wait_asynccnt(0);
#else
  asm volatile("s_wait_asynccnt 0x0" ::: "memory");
#endif
}

// ---------------------------------------------------------------------------
// Kernel 1: combined projection GEMM via V_WMMA_F32_16X16X4_F32.
// One 16x16 tile per wave, K=128 -> 32 WMMA per wave. Bias folded into pk'.
// ---------------------------------------------------------------------------
__global__ __launch_bounds__(256)
void proj_wmma_kernel(const float* __restrict__ hf,   // (4096,128)
                      const float* __restrict__ W,    // (256,64)
                      const float* __restrict__ bv,   // (64,)
                      float*       __restrict__ P) {  // (4096,128)
  const int lane = threadIdx.x & 31;
  const int wave = threadIdx.x >> 5;
  const int tile = blockIdx.x * 8 + wave;   // 2048 tiles
  const int tm   = tile >> 3;
  const int tn   = tile & 7;
  const int half = lane >> 4;
  const int l15  = lane & 15;

  const int arow = tm * 16 + l15;
  const int ncol = tn * 16 + l15;
  const int wrow_off = (tn >= 4) ? 128 : 0;
  const int wcol     = (tn >= 4) ? (ncol - 64) : ncol;

  v8f acc = {};
#pragma unroll
  for (int k0 = 0; k0 < EE; k0 += 4) {
    const int ka = k0 + 2 * half;
    v2f a, b;
    a.x = hf[arow * EE + ka];
    a.y = hf[arow * EE + ka + 1];
    b.x = W[(wrow_off + ka)     * HH + wcol];
    b.y = W[(wrow_off + ka + 1) * HH + wcol];
    acc = __builtin_amdgcn_wmma_f32_16x16x4_f32(
        false, a, false, b, (short)0, acc, false, false);
  }

  const float badd = (tn >= 4) ? bv[ncol - 64] : 0.0f;
#pragma unroll
  for (int v = 0; v < 8; ++v) {
    const int mrow = tm * 16 + v + 8 * half;
    P[mrow * EE + ncol] = acc[v] + badd;
  }
}

// ---------------------------------------------------------------------------
// Kernel 2: e + softmax for a block of 16 j-rows. 256 threads.
// Thread t owns k in {t, t+256} for all 16 rows (32 f32 accumulators).
// pk' staged async into LDS in h-chunks of 16 (512x16 f32 = 32KB), reused by
// all 16 rows -> 16x less L2 traffic; inner loop is register FMA.
// ---------------------------------------------------------------------------
__global__ __launch_bounds__(256)
void attn_softmax_kernel(const float* __restrict__ P,     // (4096,128)
                         const float* __restrict__ att,   // (64,)
                         float*       __restrict__ alpha) // (8,512,512)
{
  const int t  = threadIdx.x;
  const int b  = blockIdx.x >> 5;          // 32 j-blocks per batch
  const int j0 = (blockIdx.x & 31) * 16;

  __shared__ __align__(16) float buf[NN * 16];   // pk chunk [k][16]; later e[j][k]
  __shared__ __align__(16) float pj_s[16 * 16];  // [j][hq]
  __shared__ float att_s[HH];
  __shared__ float segr[256];

  if (t < HH) att_s[t] = att[t];

  const float* Pb  = P + (size_t)b * NN * EE;          // batch base
  const float* pjb = P + (size_t)(b * NN + j0) * EE;   // pj rows base

  float e[16][2];
#pragma unroll
  for (int j = 0; j < 16; ++j) { e[j][0] = 0.f; e[j][1] = 0.f; }

  for (int hc = 0; hc < HH; hc += 16) {
    __syncthreads();  // previous chunk fully consumed (also covers att_s)
    {   // stage pj[16][16]: one float per thread
      const int j = t >> 4, q = t & 15;
      pj_s[t] = pjb[(size_t)j * EE + hc + q];
    }
    // async-stage pk chunk: 512 rows x 16 floats = 2048 x b128 / 256 threads
#pragma unroll
    for (int i = 0; i < 8; ++i) {
      const int q = t + i * 256;          // 0..2047
      const int k = q >> 2, quad = q & 3;
      async_copy_b128(Pb + (size_t)k * EE + HH + hc + quad * 4,
                      &buf[k * 16 + quad * 4]);
    }
    wait_async0();
    __syncthreads();

    // registers: pk chunk for my 2 k's, att chunk
    v4f pva[2][4];
#pragma unroll
    for (int i = 0; i < 2; ++i) {
      const int k = t + i * 256;
#pragma unroll
      for (int c = 0; c < 4; ++c)
        pva[i][c] = *(const v4f*)&buf[k * 16 + c * 4];
    }
    float av[16];
#pragma unroll
    for (int c = 0; c < 16; ++c) av[c] = att_s[hc + c];

#pragma unroll
    for (int j = 0; j < 16; ++j) {
      v4f pjv[4];
#pragma unroll
      for (int c = 0; c < 4; ++c) pjv[c] = *(const v4f*)&pj_s[j * 16 + c * 4];
#pragma unroll
      for (int i = 0; i < 2; ++i) {
        float acc = e[j][i];
#pragma unroll
        for (int c = 0; c < 4; ++c)
#pragma unroll
          for (int q = 0; q < 4; ++q) {
            float x = pjv[c][q] + pva[i][c][q];
            x = (x > 0.f) ? x : 0.2f * x;               // leaky_relu 0.2
            acc = __builtin_fmaf(av[c * 4 + q], x, acc);
          }
        e[j][i] = acc;
      }
    }
  }

  // ---- softmax over k per row; e matrix -> LDS (reuse buf: 16x512 f32) ----
  __syncthreads();
#pragma unroll
  for (int j = 0; j < 16; ++j) {
    buf[j * NN + t]       = e[j][0];
    buf[j * NN + t + 256] = e[j][1];
  }
  __syncthreads();

  const int jr = t >> 4;        // my row
  const int sg = t & 15;        // my 32-wide segment
  float* row = &buf[jr * NN + sg * 32];

  float m = -3.402823466e38f;
#pragma unroll
  for (int q = 0; q < 32; ++q) m = fmaxf(m, row[q]);
  segr[t] = m;
  __syncthreads();
  if (sg == 0) {
    float mm = segr[jr * 16];
#pragma unroll
    for (int q = 1; q < 16; ++q) mm = fmaxf(mm, segr[jr * 16 + q]);
    segr[jr * 16] = mm;
  }
  __syncthreads();
  const float rowm = segr[jr * 16];
  __syncthreads();

  float s = 0.f;
#pragma unroll
  for (int q = 0; q < 32; ++q) {
    const float p = __expf(row[q] - rowm);
    row[q] = p;
    s += p;
  }
  segr[t] = s;
  __syncthreads();
  if (sg == 0) {
    float ss = 0.f;
#pragma unroll
    for (int q = 0; q < 16; ++q) ss += segr[jr * 16 + q];
    segr[jr * 16] = ss;
  }
  __syncthreads();
  const float inv = 1.0f / segr[jr * 16];

  // coalesced b128 stores: row (b, j0+jr), 32 consecutive floats per thread
  float* arow = alpha + (size_t)(b * NN + j0 + jr) * NN + sg * 32;
#pragma unroll
  for (int c = 0; c < 8; ++c) {
    v4f vo;
#pragma unroll
    for (int q = 0; q < 4; ++q) vo[q] = row[c * 4 + q] * inv;
    *(v4f*)(arow + c * 4) = vo;
  }
}

// ---------------------------------------------------------------------------
// Kernel 3: edge_index (int64) + edge_weights, one thread per edge.
// ---------------------------------------------------------------------------
__global__ __launch_bounds__(256)
void edge_kernel(const float* __restrict__ alpha,
                 long long*   __restrict__ ei,   // (2, BE)
                 float*       __restrict__ ew) { // (BE,)
  const long long gid = (long long)blockIdx.x * 256 + threadIdx.x;
  if (gid >= (long long)BE) return;
  const int b = (int)(gid / BE1);
  const int r = (int)(gid - (long long)b * BE1);
  const int s = r / (NN - 1);
  const int p = r - s * (NN - 1);
  const int d = p + (p >= s ? 1 : 0);
  const long long off = (long long)b * NN;
  ei[gid]                 = off + s;
  ei[gid + (long long)BE] = off + d;
  const size_t base = (size_t)b * NN * NN;
  ew[gid] = 0.5f * (alpha[base + (size_t)s * NN + d] +
                    alpha[base + (size_t)d * NN + s]);
}

// ---------------------------------------------------------------------------
extern "C" void kernel_launch(void* const* d_in, const int* in_sizes, int n_in,
                              void* d_out, int out_size, void* d_ws, size_t ws_size,
                              hipStream_t stream) {
  const float* hf  = (const float*)d_in[0];  // (4096,128)
  const float* W   = (const float*)d_in[1];  // (256,64)
  const float* bv  = (const float*)d_in[2];  // (64,)
  const float* att = (const float*)d_in[3];  // (64,)

  float* P = (float*)d_ws;                   // 4096x128 f32 = 2MB scratch

  // d_out: edge_index int64[2*BE] | edge_weights f32[BE] | alpha f32[8*512*512]
  long long* ei    = (long long*)d_out;
  float*     ew    = (float*)(ei + 2LL * BE);
  float*     alpha = ew + BE;

  proj_wmma_kernel<<<256, 256, 0, stream>>>(hf, W, bv, P);
  attn_softmax_kernel<<<BB * (NN / 16), 256, 0, stream>>>(P, att, alpha);
  edge_kernel<<<(BE + 255) / 256, 256, 0, stream>>>(alpha, ei, ew);
}